// SingleSigmaKernel_79439715107058
// MI455X (gfx1250) — compile-verified
//
#include <hip/hip_runtime.h>

// Gaussian-kernel MMD loss on MI455X (gfx1250).
// d2 for a 16x16 tile of pairs is computed EXACTLY in f32 by one
// V_WMMA_F32_16X16X4_F32 using A row = [x, y, x^2+y^2, 1],
// B col = [-2x', -2y', 1, x'^2+y'^2]  =>  A.B = ||p - q||^2.

typedef float v2f __attribute__((ext_vector_type(2)));
typedef float v8f __attribute__((ext_vector_type(8)));

#define TILE 16
#define PREP_THREADS 256
#define JCHUNKS 8
#define LOG2E 1.4426950408889634f

// ---- Prep: scale by exp(log_scale), subtract centroid (plain mean) -------------
__global__ void prep_center_kernel(const float* __restrict__ pts,      // [N,2]
                                   const float* __restrict__ log_scale,// [2]
                                   float2* __restrict__ out, int N) {
  __shared__ float sX[PREP_THREADS];
  __shared__ float sY[PREP_THREADS];
  const float sx = __builtin_amdgcn_exp2f(log_scale[0] * LOG2E);
  const float sy = __builtin_amdgcn_exp2f(log_scale[1] * LOG2E);
  const int tid = threadIdx.x;
  float ax = 0.0f, ay = 0.0f;
  for (int i = tid; i < N; i += PREP_THREADS) {
    ax += pts[2 * i] * sx;
    ay += pts[2 * i + 1] * sy;
  }
  sX[tid] = ax;
  sY[tid] = ay;
  __syncthreads();
  for (int off = PREP_THREADS / 2; off > 0; off >>= 1) {
    if (tid < off) { sX[tid] += sX[tid + off]; sY[tid] += sY[tid + off]; }
    __syncthreads();
  }
  const float mx = sX[0] / (float)N;
  const float my = sY[0] / (float)N;
  for (int i = tid; i < N; i += PREP_THREADS) {
    float2 v;
    v.x = pts[2 * i] * sx - mx;
    v.y = pts[2 * i + 1] * sy - my;
    out[i] = v;
  }
}

// ---- Main: WMMA tile d^2 + exp2 accumulation -------------------------------
__global__ void mmd_wmma_kernel(const float2* __restrict__ baseC,
                                const float2* __restrict__ targetC,
                                const float* __restrict__ log_sigma,
                                float* __restrict__ partials,
                                int N, int NT, int totalWaves) {
  const int lane = threadIdx.x & 31;
  // Wave id is uniform per wave: force it into an SGPR so all derived
  // decode values / loop bounds are scalar (s_cbranch loops, no exec juggling).
  const int gwave =
      __builtin_amdgcn_readfirstlane((int)((blockIdx.x * blockDim.x + threadIdx.x) >> 5));
  if (gwave >= totalWaves) return;  // wave-uniform guard

  const int perTerm = NT * JCHUNKS;
  const int term = gwave / perTerm;
  const int rem = gwave % perTerm;
  const int it = rem / JCHUNKS;
  const int chunk = rem % JCHUNKS;
  const int jspan = NT / JCHUNKS;
  const int jBegin = chunk * jspan;
  const int jEnd = jBegin + jspan;

  const float invN2 = 1.0f / ((float)N * (float)N);
  const float2* P;
  const float2* Q;
  float coef;
  if (term == 0)      { P = baseC;   Q = baseC;   coef = invN2; }
  else if (term == 1) { P = targetC; Q = targetC; coef = invN2; }
  else                { P = baseC;   Q = targetC; coef = -2.0f * invN2; }

  // exponent coefficient: exp(-d2 * inv2s2) = exp2(d2 * c2),  c2 < 0 always
  const float ls = log_sigma[0];
  const float inv2s2 = 0.5f * __builtin_amdgcn_exp2f(-2.0f * ls * LOG2E);
  const float c2 = -inv2s2 * LOG2E;

  // A fragment (16x4 f32, 2 VGPRs): lanes 0-15 -> K={0,1}, lanes 16-31 -> K={2,3}
  const float2 p = P[it * TILE + (lane & 15)];
  v2f a;
  if (lane < 16) { a[0] = p.x; a[1] = p.y; }
  else           { a[0] = p.x * p.x + p.y * p.y; a[1] = 1.0f; }

  float sum = 0.0f;
#pragma unroll 2
  for (int jt = jBegin; jt < jEnd; ++jt) {
    // B fragment (4x16 f32, 2 VGPRs): lanes 0-15 -> rows {0,1}, 16-31 -> rows {2,3}
    const float2 q = Q[jt * TILE + (lane & 15)];
    v2f b;
    if (lane < 16) { b[0] = -2.0f * q.x; b[1] = -2.0f * q.y; }
    else           { b[0] = 1.0f;        b[1] = q.x * q.x + q.y * q.y; }

    v8f c = {};
    // D = A x B (+0): full-tile exact squared distances in f32
    v8f d = __builtin_amdgcn_wmma_f32_16x16x4_f32(
        /*neg_a=*/false, a, /*neg_b=*/false, b,
        /*c_mod=*/(short)0, c, /*reuse_a=*/false, /*reuse_b=*/false);

#pragma unroll
    for (int k = 0; k < 8; ++k) {
      // c2 < 0, so min(d2*c2, 0) == max(d2,0)*c2  (matches jnp.maximum(..,0))
      // and avoids the extra canonicalize+clamp VALU ops on the WMMA output.
      const float t = fminf(d[k] * c2, 0.0f);
      sum += __builtin_amdgcn_exp2f(t);  // v_exp_f32 (TRANS, co-executes)
    }
  }

  sum *= coef;
  // wave32 butterfly reduction
  for (int off = 16; off > 0; off >>= 1) sum += __shfl_xor(sum, off, 32);
  if (lane == 0) partials[gwave] = sum;
}

// ---- Final deterministic reduction ----------------------------------------
__global__ void reduce_kernel(const float* __restrict__ partials, int n,
                              float* __restrict__ out) {
  __shared__ float s[256];
  float a = 0.0f;
  for (int i = threadIdx.x; i < n; i += 256) a += partials[i];
  s[threadIdx.x] = a;
  __syncthreads();
  for (int off = 128; off > 0; off >>= 1) {
    if ((int)threadIdx.x < off) s[threadIdx.x] += s[threadIdx.x + off];
    __syncthreads();
  }
  if (threadIdx.x == 0) out[0] = s[0];
}

extern "C" void kernel_launch(void* const* d_in, const int* in_sizes, int n_in,
                              void* d_out, int out_size, void* d_ws, size_t ws_size,
                              hipStream_t stream) {
  const float* base      = (const float*)d_in[0];  // [N,2]
  const float* target    = (const float*)d_in[1];  // [M,2] (M == N here)
  const float* log_sigma = (const float*)d_in[2];  // scalar
  const float* log_scale = (const float*)d_in[3];  // [2]

  const int N = in_sizes[0] / 2;   // 8192
  const int NT = N / TILE;         // 512 tiles per dimension
  const int totalWaves = 3 * NT * JCHUNKS;          // 12288
  const int blocks = (totalWaves * 32 + 255) / 256; // 1536

  // Workspace layout: baseC [N float2] | targetC [N float2] | partials [totalWaves]
  float2* baseC   = (float2*)d_ws;
  float2* targetC = baseC + N;
  float*  partials = (float*)(targetC + N);

  prep_center_kernel<<<1, PREP_THREADS, 0, stream>>>(base, log_scale, baseC, N);
  prep_center_kernel<<<1, PREP_THREADS, 0, stream>>>(target, log_scale, targetC, N);
  mmd_wmma_kernel<<<blocks, 256, 0, stream>>>(baseC, targetC, log_sigma,
                                              partials, N, NT, totalWaves);
  reduce_kernel<<<1, 256, 0, stream>>>(partials, totalWaves, (float*)d_out);
}